// AuxLossFreeRouter_35150012351315
// MI455X (gfx1250) — compile-verified
//
#include <hip/hip_runtime.h>
#include <math.h>

#define NUM_EXPERTS 64
#define TOP_K 8
#define WAVES_PER_BLOCK 4
#define TOKENS_PER_WAVE 32
#define TOKENS_PER_BLOCK (WAVES_PER_BLOCK * TOKENS_PER_WAVE)  // 128

typedef __attribute__((ext_vector_type(2))) float v2f;
typedef __attribute__((ext_vector_type(4))) float v4f;
typedef __attribute__((ext_vector_type(8))) float v8f;

// One WMMA k-step: HALF selects which v4f of the 8-float group, PAIR selects
// the adjacent float pair. Both must be literal constants (shufflevector
// requires frontend-constant indices).
#define WMMA_STEP(HALF, PAIR)                                                   \
    {                                                                           \
        const v2f a0 = __builtin_shufflevector(aA[0][HALF], aA[0][HALF],        \
                                               2 * (PAIR), 2 * (PAIR) + 1);     \
        const v2f a1 = __builtin_shufflevector(aA[1][HALF], aA[1][HALF],        \
                                               2 * (PAIR), 2 * (PAIR) + 1);     \
        _Pragma("unroll")                                                       \
        for (int t = 0; t < 4; ++t) {                                           \
            const v2f bf = __builtin_shufflevector(bB[t][HALF], bB[t][HALF],    \
                                                   2 * (PAIR), 2 * (PAIR) + 1); \
            acc[0][t] = __builtin_amdgcn_wmma_f32_16x16x4_f32(                  \
                false, a0, false, bf, (short)0, acc[0][t], false, false);       \
            acc[1][t] = __builtin_amdgcn_wmma_f32_16x16x4_f32(                  \
                false, a1, false, bf, (short)0, acc[1][t], false, false);       \
        }                                                                       \
    }

// ---------------------------------------------------------------------------
// Kernel 1: adaptive bias computation + workspace accumulator reset.
// ws layout (floats): [0:64) batch-load accumulator (zeroed here)
//                     [64:128) bias vector
// ---------------------------------------------------------------------------
__global__ __launch_bounds__(64) void router_bias_kernel(
    const float* __restrict__ expert_loads,
    const float* __restrict__ bias_strength,
    float* __restrict__ ws,
    float* __restrict__ out_bs)
{
    __shared__ float sh[NUM_EXPERTS];
    const int e = threadIdx.x;
    const float le = expert_loads[e];
    sh[e] = le;
    __syncthreads();
    float S = 0.0f;
    for (int i = 0; i < NUM_EXPERTS; ++i) S += sh[i];
    S = fmaxf(S, 1e-8f);
    const float q = le / S;
    const float t = 1.0f / (float)NUM_EXPERTS;
    const float term = t * (__logf(t) - __logf(fmaxf(q, 1e-8f)));
    __syncthreads();
    sh[e] = term;
    __syncthreads();
    float kl = 0.0f;
    for (int i = 0; i < NUM_EXPERTS; ++i) kl += sh[i];
    const float adaptive = 1.0f / (1.0f + __expf(-10.0f * kl));
    const float nbs = 0.9f * bias_strength[0] + 0.1f * adaptive;
    ws[e] = 0.0f;  // reset batch-load accumulator every launch (determinism)
    ws[NUM_EXPERTS + e] = tanhf((q - t) * (float)NUM_EXPERTS) * nbs;
    if (e == 0) out_bs[0] = nbs;
}

// ---------------------------------------------------------------------------
// Kernel 2: WMMA router GEMM + noise/bias + top-8 + softmax + load scatter.
// One wave -> 32 tokens x 64 experts (2 A-tiles x 4 B-tiles of
// V_WMMA_F32_16X16X4_F32). K is block-permuted per 16-chunk so every lane's
// fragment data for 4 consecutive WMMA steps is 8 contiguous floats
// (two aligned b128 loads). A/B share the permutation -> result unchanged.
// ---------------------------------------------------------------------------
__global__ __launch_bounds__(128) void router_main_kernel(
    const float* __restrict__ x,      // [N, K] tokens
    const float* __restrict__ w,      // [64, K] router weights
    const float* __restrict__ noise,  // [N, 64]
    float* __restrict__ ws,           // [0:64) acc, [64:128) bias
    float* __restrict__ out_idx,      // [N, 8] (indices as float)
    float* __restrict__ out_w,        // [N, 8]
    int K)
{
    __shared__ float s_log[WAVES_PER_BLOCK * TOKENS_PER_WAVE * NUM_EXPERTS];  // 32KB
    __shared__ float s_bias[NUM_EXPERTS];
    __shared__ float s_acc[NUM_EXPERTS];

    const int tid  = threadIdx.x;
    const int wave = tid >> 5;
    const int lane = tid & 31;
    const int g    = lane >> 4;   // half-wave: owns k-subblock 8g..8g+7 of each 16-chunk
    const int nl   = lane & 15;   // row (A) / column (B,C) index

    if (tid < NUM_EXPERTS) {
        s_bias[tid] = ws[NUM_EXPERTS + tid];
        s_acc[tid]  = 0.0f;
    }
    __syncthreads();

    const int tok0 = blockIdx.x * TOKENS_PER_BLOCK + wave * TOKENS_PER_WAVE;

    // ---- GEMM: 2 token tiles x 4 expert tiles over K ----
    v8f acc[2][4] = {};
    const float* xp0 = x + (size_t)(tok0 + nl) * (size_t)K + 8 * g;
    const float* xp1 = xp0 + (size_t)16 * (size_t)K;
    const float* wp  = w + (size_t)nl * (size_t)K + 8 * g;

#pragma unroll 2
    for (int kk = 0; kk < K; kk += 16) {
        v4f aA[2][2], bB[4][2];
        aA[0][0] = *(const v4f*)(xp0 + kk);
        aA[0][1] = *(const v4f*)(xp0 + kk + 4);
        aA[1][0] = *(const v4f*)(xp1 + kk);
        aA[1][1] = *(const v4f*)(xp1 + kk + 4);
#pragma unroll
        for (int t = 0; t < 4; ++t) {
            bB[t][0] = *(const v4f*)(wp + t * 16 * K + kk);
            bB[t][1] = *(const v4f*)(wp + t * 16 * K + kk + 4);
        }
        WMMA_STEP(0, 0)
        WMMA_STEP(0, 1)
        WMMA_STEP(1, 0)
        WMMA_STEP(1, 1)
    }

    // ---- Dump C layout to LDS, folding noise*0.01 - bias[e] ----
#pragma unroll
    for (int m = 0; m < 2; ++m) {
#pragma unroll
        for (int t = 0; t < 4; ++t) {
            const int col = t * 16 + nl;
            const float bcol = s_bias[col];
#pragma unroll
            for (int r = 0; r < 8; ++r) {
                const int row = m * 16 + 8 * g + r;  // C: VGPR r -> M = 8g+r
                const float val = acc[m][t][r]
                    + 0.01f * noise[(size_t)(tok0 + row) * NUM_EXPERTS + col]
                    - bcol;
                s_log[(wave * TOKENS_PER_WAVE + row) * NUM_EXPERTS + col] = val;
            }
        }
    }
    __syncthreads();

    // ---- Top-8 (first-index tie-break, like lax.top_k) + softmax ----
    // All 32 lanes active: one token per lane.
    {
        const int gtok = tok0 + lane;
        const float* lv = &s_log[(wave * TOKENS_PER_WAVE + lane) * NUM_EXPERTS];
        int   sel[TOP_K];
        float selv[TOP_K];
#pragma unroll
        for (int kp = 0; kp < TOP_K; ++kp) {
            float best = -INFINITY;
            int bidx = 0;
            for (int e = 0; e < NUM_EXPERTS; ++e) {
                bool taken = false;
#pragma unroll
                for (int j = 0; j < TOP_K; ++j)
                    taken = taken || (j < kp && sel[j] == e);
                const float v = lv[e];
                if (!taken && v > best) { best = v; bidx = e; }
            }
            sel[kp] = bidx;
            selv[kp] = best;
        }
        // softmax over the 8 selected (selv[0] is the max: sorted descending)
        float wgt[TOP_K];
        float sum = 0.0f;
#pragma unroll
        for (int j = 0; j < TOP_K; ++j) {
            wgt[j] = __expf(selv[j] - selv[0]);
            sum += wgt[j];
        }
        const float inv = 1.0f / sum;
#pragma unroll
        for (int j = 0; j < TOP_K; ++j) {
            const float ww = wgt[j] * inv;
            out_idx[(size_t)gtok * TOP_K + j] = (float)sel[j];
            out_w[(size_t)gtok * TOP_K + j]   = ww;
            atomicAdd(&s_acc[sel[j]], ww);
        }
    }
    __syncthreads();

    if (tid < NUM_EXPERTS) atomicAdd(&ws[tid], s_acc[tid]);
}

// ---------------------------------------------------------------------------
// Kernel 3: EMA load update.
// ---------------------------------------------------------------------------
__global__ __launch_bounds__(64) void router_finalize_kernel(
    const float* __restrict__ expert_loads,
    const float* __restrict__ ws,
    float* __restrict__ out_loads,
    float inv_n)
{
    const int e = threadIdx.x;
    out_loads[e] = 0.999f * expert_loads[e] + 0.001f * (ws[e] * inv_n);
}

// ---------------------------------------------------------------------------
extern "C" void kernel_launch(void* const* d_in, const int* in_sizes, int n_in,
                              void* d_out, int out_size, void* d_ws, size_t ws_size,
                              hipStream_t stream) {
    const float* x     = (const float*)d_in[0];  // hidden_states [4,4096,2048]
    const float* rw    = (const float*)d_in[1];  // router_w [64,2048]
    const float* loads = (const float*)d_in[2];  // expert_loads [64]
    const float* bs    = (const float*)d_in[3];  // bias_strength [1]
    const float* noise = (const float*)d_in[4];  // noise [N,64]

    const int N = in_sizes[4] / NUM_EXPERTS;     // 16384 tokens
    const int K = in_sizes[1] / NUM_EXPERTS;     // 2048 hidden dim

    float* out       = (float*)d_out;
    float* out_idx   = out;                               // [N,8]
    float* out_w     = out + (size_t)N * TOP_K;           // [N,8]
    float* out_loads = out + (size_t)N * TOP_K * 2;       // [64]
    float* out_bs    = out_loads + NUM_EXPERTS;           // [1]
    float* ws        = (float*)d_ws;

    router_bias_kernel<<<1, 64, 0, stream>>>(loads, bs, ws, out_bs);

    const int blocks = (N + TOKENS_PER_BLOCK - 1) / TOKENS_PER_BLOCK;
    router_main_kernel<<<blocks, 128, 0, stream>>>(x, rw, noise, ws,
                                                   out_idx, out_w, K);

    router_finalize_kernel<<<1, 64, 0, stream>>>(loads, ws, out_loads,
                                                 1.0f / (float)N);
}